// SSMKernelDPLR_14980845928686
// MI455X (gfx1250) — compile-verified
//
#include <hip/hip_runtime.h>

// ---------------------------------------------------------------------------
// S4-DPLR convolution kernel for MI455X (gfx1250, wave32, WMMA).
// Stage 1: Cauchy resolvent + rank-1 Woodbury -> frequency response (H x Lf)
// Stage 2: irfft as a real DFT GEMM using V_WMMA_F32_16X16X4_F32, with the
//          cos/sin basis gathered from 16KB LDS tables (basis is periodic in
//          (l*t) mod L, so it is never materialized in memory).
// ---------------------------------------------------------------------------

typedef float v2f __attribute__((ext_vector_type(2)));
typedef float v8f __attribute__((ext_vector_type(8)));

#define HDIM   1024
#define NSTATE 32
#define LCONV  4096
#define LF     2049      // L/2 + 1
#define KPAD   2052      // LF rounded up to multiple of 4 (WMMA K-step)
#define PI_F   3.14159265358979323846f

// ===========================================================================
// Stage 1: per-(h,l) frequency response.
//   z_l     = 2i * tan(pi*l/L)            (bilinear image of FFT node)
//   2/(1+w) = 1 + i * tan(pi*l/L)         (closed form, no 1/0 at Nyquist)
//   r_ab(l) = dt * sum_n [ v_ab/(z - w) + conj(v_ab)/(z - conj(w)) ]
//   k_f     = (r00 - r01*r10/(1+r11)) * (1 + i t)
//   Kc = s_l*Re(k_f), Ks = -s_l*Im(k_f),  s_l = irfft weight (1 or 2)/L
// ===========================================================================
__global__ void __launch_bounds__(256)
s4d_freq_kernel(const float* __restrict__ log_dt,
                const float* __restrict__ A_re, const float* __restrict__ A_im,
                const float* __restrict__ B_re, const float* __restrict__ B_im,
                const float* __restrict__ C_re, const float* __restrict__ C_im,
                const float* __restrict__ P_re, const float* __restrict__ P_im,
                float* __restrict__ Kc, float* __restrict__ Ks)
{
    __shared__ float sw_re[NSTATE], sw_im[NSTATE];
    __shared__ float s00r[NSTATE], s00i[NSTATE];
    __shared__ float s01r[NSTATE], s01i[NSTATE];
    __shared__ float s10r[NSTATE], s10i[NSTATE];
    __shared__ float s11r[NSTATE], s11i[NSTATE];

    const int   h  = blockIdx.x;
    const float dt = __expf(log_dt[h]);

    if (threadIdx.x < NSTATE) {
        const int n = threadIdx.x;
        const int o = h * NSTATE + n;
        const float ar = A_re[o], ai = A_im[o];
        const float br = B_re[o], bi = B_im[o];
        const float cr = C_re[o], ci = C_im[o];
        const float pr = P_re[o], pi = P_im[o];
        sw_re[n] = ar * dt;  sw_im[n] = ai * dt;        // discretized pole
        s00r[n] = br*cr - bi*ci;  s00i[n] = br*ci + bi*cr;   // B*C
        s01r[n] = br*pr + bi*pi;  s01i[n] = bi*pr - br*pi;   // B*conj(P)
        s10r[n] = pr*cr - pi*ci;  s10i[n] = pr*ci + pi*cr;   // P*C
        s11r[n] = pr*pr + pi*pi;  s11i[n] = 0.0f;            // |P|^2
    }
    __syncthreads();

    for (int l = threadIdx.x; l < KPAD; l += blockDim.x) {
        float outc = 0.0f, outs = 0.0f;
        if (l < LF) {
            const float t  = __tanf(PI_F * (float)l / (float)LCONV);
            const float z2 = 2.0f * t;                  // z = i*z2

            float a00r=0,a00i=0,a01r=0,a01i=0,a10r=0,a10i=0,a11r=0,a11i=0;
            #pragma unroll 4
            for (int n = 0; n < NSTATE; ++n) {
                const float wr = sw_re[n], wi = sw_im[n];
                const float xr  = -wr;
                const float y1  = z2 - wi;
                const float y2  = z2 + wi;
                const float iv1 = 1.0f / (xr*xr + y1*y1);
                const float iv2 = 1.0f / (xr*xr + y2*y2);
                const float d1r =  xr*iv1, d1i = -y1*iv1;   // 1/(z - w)
                const float d2r =  xr*iv2, d2i = -y2*iv2;   // 1/(z - conj(w))
                const float sr  = d1r + d2r, dri = d1i - d2i;
                const float si  = d1i + d2i, drr = d1r - d2r;
                // acc += v*d1 + conj(v)*d2
                a00r += s00r[n]*sr - s00i[n]*dri;  a00i += s00r[n]*si + s00i[n]*drr;
                a01r += s01r[n]*sr - s01i[n]*dri;  a01i += s01r[n]*si + s01i[n]*drr;
                a10r += s10r[n]*sr - s10i[n]*dri;  a10i += s10r[n]*si + s10i[n]*drr;
                a11r += s11r[n]*sr - s11i[n]*dri;  a11i += s11r[n]*si + s11i[n]*drr;
            }
            const float r00r=dt*a00r, r00i=dt*a00i;
            const float r01r=dt*a01r, r01i=dt*a01i;
            const float r10r=dt*a10r, r10i=dt*a10i;
            const float r11r=dt*a11r, r11i=dt*a11i;

            // Woodbury: k_f = r00 - r01*r10 / (1 + r11)
            const float dr  = 1.0f + r11r, di = r11i;
            const float idn = 1.0f / (dr*dr + di*di);
            const float nr  = r01r*r10r - r01i*r10i;
            const float ni  = r01r*r10i + r01i*r10r;
            const float kfr = r00r - (nr*dr + ni*di) * idn;
            const float kfi = r00i - (ni*dr - nr*di) * idn;

            // bilinear factor (1 + i t), then irfft weight
            const float fr = kfr - kfi * t;
            const float fi = kfi + kfr * t;
            const float s  = (l == 0 || l == LCONV/2) ? (1.0f/LCONV) : (2.0f/LCONV);
            outc =  s * fr;
            outs = -s * fi;
        }
        Kc[(size_t)h * KPAD + l] = outc;
        Ks[(size_t)h * KPAD + l] = outs;
    }
}

// ===========================================================================
// Stage 2: k[h,t] = sum_l Kc[h,l]*cos(2*pi*l*t/L) + Ks[h,l]*sin(2*pi*l*t/L)
// as an f32 WMMA GEMM. Each wave32 owns a 16(h) x 64(t) output tile:
// 4 accumulators, 8 v_wmma_f32_16x16x4_f32 per K-step (cos + sin into the
// same C), 513 K-steps. Basis values gathered from LDS tables with a modular
// index recurrence: idx(l,t) = (l*t) mod 4096.
// ===========================================================================
__global__ void __launch_bounds__(256)
irdft_wmma_kernel(const float* __restrict__ Kc,
                  const float* __restrict__ Ks,
                  float* __restrict__ out)
{
    __shared__ float ctab[LCONV];
    __shared__ float stab[LCONV];
    for (int j = threadIdx.x; j < LCONV; j += blockDim.x) {
        float s, c;
        __sincosf((2.0f * PI_F / (float)LCONV) * (float)j, &s, &c);
        ctab[j] = c;
        stab[j] = s;
    }
    __syncthreads();

    const int lane  = threadIdx.x & 31;
    const int wave  = threadIdx.x >> 5;
    const int gw    = blockIdx.x * 8 + wave;   // 4096 waves total
    const int htile = gw >> 6;                 // 0..63  -> rows h0..h0+15
    const int tblk  = gw & 63;                 // 0..63  -> cols t0..t0+63
    const int h0    = htile << 4;
    const int t0    = tblk  << 6;

    const int m  = lane & 15;                  // row (A) / col (B) within tile
    const int dl = (lane >> 4) << 1;           // K sub-offset: 0 or 2

    v8f acc[4];
    #pragma unroll
    for (int nt = 0; nt < 4; ++nt)
        #pragma unroll
        for (int i = 0; i < 8; ++i) acc[nt][i] = 0.0f;

    int tval[4], dlt[4], t4[4], base[4];
    #pragma unroll
    for (int nt = 0; nt < 4; ++nt) {
        const int t = t0 + nt * 16 + m;
        tval[nt] = t;
        dlt[nt]  = (dl * t) & (LCONV - 1);
        t4[nt]   = (4  * t) & (LCONV - 1);
        base[nt] = 0;                          // (k0 * t) mod L, k0 = 0
    }

    // A operand rows: lane m holds row h0+m, K pair starting at k0+dl.
    const float* arow = Kc + (size_t)(h0 + m) * KPAD + dl;
    const float* brow = Ks + (size_t)(h0 + m) * KPAD + dl;

    for (int k0 = 0; k0 < KPAD; k0 += 4) {
        const v2f Ac = *(const v2f*)(arow + k0);   // 8B aligned: row*KPAD+dl+k0 even
        const v2f As = *(const v2f*)(brow + k0);
        #pragma unroll
        for (int nt = 0; nt < 4; ++nt) {
            const int j0 = (base[nt] + dlt[nt]) & (LCONV - 1);
            const int j1 = (j0 + tval[nt])      & (LCONV - 1);
            v2f Bc = { ctab[j0], ctab[j1] };
            v2f Bs = { stab[j0], stab[j1] };
            acc[nt] = __builtin_amdgcn_wmma_f32_16x16x4_f32(
                false, Ac, false, Bc, (short)0, acc[nt], false, false);
            acc[nt] = __builtin_amdgcn_wmma_f32_16x16x4_f32(
                false, As, false, Bs, (short)0, acc[nt], false, false);
            base[nt] = (base[nt] + t4[nt]) & (LCONV - 1);
        }
    }

    // D layout: VGPR r -> row (lane<16 ? r : r+8), col = lane&15 of the tile.
    const int rbase = (lane >> 4) << 3;
    #pragma unroll
    for (int nt = 0; nt < 4; ++nt) {
        const int tcol = t0 + nt * 16 + m;
        #pragma unroll
        for (int r = 0; r < 8; ++r)
            out[(size_t)(h0 + rbase + r) * LCONV + tcol] = acc[nt][r];
    }
}

// ===========================================================================
extern "C" void kernel_launch(void* const* d_in, const int* in_sizes, int n_in,
                              void* d_out, int out_size, void* d_ws, size_t ws_size,
                              hipStream_t stream)
{
    (void)in_sizes; (void)n_in; (void)out_size; (void)ws_size;

    const float* log_dt = (const float*)d_in[0];
    const float* A_re   = (const float*)d_in[1];
    const float* A_im   = (const float*)d_in[2];
    const float* B_re   = (const float*)d_in[3];
    const float* B_im   = (const float*)d_in[4];
    const float* C_re   = (const float*)d_in[5];
    const float* C_im   = (const float*)d_in[6];
    const float* P_re   = (const float*)d_in[7];
    const float* P_im   = (const float*)d_in[8];
    // d_in[9] = L (scalar, == 4096, compiled in as LCONV)

    float* Kc = (float*)d_ws;                           // H x KPAD
    float* Ks = Kc + (size_t)HDIM * KPAD;               // H x KPAD (~16.8 MB total)

    s4d_freq_kernel<<<dim3(HDIM), dim3(256), 0, stream>>>(
        log_dt, A_re, A_im, B_re, B_im, C_re, C_im, P_re, P_im, Kc, Ks);

    // 4096 waves = 512 blocks x 8 waves; each wave -> 16x64 output tile.
    irdft_wmma_kernel<<<dim3(512), dim3(256), 0, stream>>>(
        Kc, Ks, (float*)d_out);
}